// Linear5_58875411693678
// MI455X (gfx1250) — compile-verified
//
#include <hip/hip_runtime.h>
#include <math.h>

#define KT 96     // trials
#define DD 256    // embed dim
#define PB 40     // phoneme bins
#define NROWS (KT * PB)   // 3840
#define CHUNK 128         // K-chunk staged through LDS
#define LDSTR 132         // LDS row stride (floats): 132%64==4 -> conflict-free fragment reads

typedef float v2f __attribute__((ext_vector_type(2)));
typedef float v8f __attribute__((ext_vector_type(8)));

// ---------------------------------------------------------------------------
// Kernel 1: L2-normalize each column e[k,:,i] over D and write row-major
// A[r=k*PB+i][d] (D-contiguous) into workspace. Deterministic tree reduction.
// ---------------------------------------------------------------------------
__global__ __launch_bounds__(DD)
void normalize_cols(const float* __restrict__ enroll,
                    const float* __restrict__ test,
                    float* __restrict__ A, float* __restrict__ B) {
  __shared__ float red[DD];
  const int b = blockIdx.x;                       // 0 .. 2*NROWS-1
  const float* src = (b < NROWS) ? enroll : test;
  float* dst       = (b < NROWS) ? A      : B;
  const int r = (b < NROWS) ? b : (b - NROWS);
  const int k = r / PB, i = r % PB;
  const int d = threadIdx.x;

  const float v = src[(size_t)(k * DD + d) * PB + i];
  red[d] = v * v;
  __syncthreads();
  for (int s = DD / 2; s > 0; s >>= 1) {
    if (d < s) red[d] += red[d + s];
    __syncthreads();
  }
  const float n = fmaxf(sqrtf(red[0]), 1e-12f);
  dst[(size_t)r * DD + d] = v / n;
}

// tiny 1->8->1 gate MLP: sum_c w2[c] * tanh(x*w1[c] + b1[c])
__device__ __forceinline__ float gate8(float x, const float* __restrict__ w1,
                                       const float* __restrict__ b1,
                                       const float* __restrict__ w2) {
  float g = 0.f;
#pragma unroll
  for (int c = 0; c < 8; ++c) g += w2[c] * tanhf(fmaf(x, w1[c], b1[c]));
  return g;
}

// ---------------------------------------------------------------------------
// Kernel 2: one block per (k,l) trial pair. 9 waves, each owns one 16x16 WMMA
// tile of the padded 48x48 pho tile. f32 WMMA over K=256 in two LDS chunks,
// then gate -> masked row reduction (deterministic shuffle tree) -> gate ->
// masked final reduction -> out[k*96+l].
// ---------------------------------------------------------------------------
__global__ __launch_bounds__(288)
void pho_fused(const float* __restrict__ A, const float* __restrict__ B,
               const float* __restrict__ fc1w, const float* __restrict__ fc1b,
               const float* __restrict__ fc2w, float* __restrict__ out) {
  __shared__ float eT[PB * LDSTR];   // 40 x 128 chunk of e-rows (k-trial)
  __shared__ float tT[PB * LDSTR];   // 40 x 128 chunk of t-rows (l-trial)
  __shared__ float pS[48][3];        // per-row gated partial sums (3 j-tiles)
  __shared__ float pC[48][3];        // per-row nonzero partial counts
  __shared__ float s2S[PB];
  __shared__ float s2C[PB];

  const int kTrial = blockIdx.x / KT;
  const int lTrial = blockIdx.x % KT;
  const int tid  = threadIdx.x;      // 0..287
  const int wave = tid >> 5;         // 0..8 (wave32)
  const int lane = tid & 31;

  float w1[8], b1[8], w2[8];
#pragma unroll
  for (int c = 0; c < 8; ++c) { w1[c] = fc1w[c]; b1[c] = fc1b[c]; w2[c] = fc2w[c]; }

  const float* gA = A + (size_t)kTrial * PB * DD;
  const float* gB = B + (size_t)lTrial * PB * DD;

  // fragment addressing (ISA 16x4 f32 A layout): lane L -> row L&15,
  // k-subcolumn 2*(L>>4); rows >= 40 clamped (results excluded later).
  const int tm = wave / 3, tn = wave % 3;
  const int rowHalf = lane & 15;
  const int kk = (lane >> 4) * 2;
  int ar = tm * 16 + rowHalf; if (ar >= PB) ar = PB - 1;
  int br = tn * 16 + rowHalf; if (br >= PB) br = PB - 1;
  const float* aBase = &eT[ar * LDSTR + kk];
  const float* bBase = &tT[br * LDSTR + kk];

  v8f acc = {0.f, 0.f, 0.f, 0.f, 0.f, 0.f, 0.f, 0.f};

  for (int kc = 0; kc < DD; kc += CHUNK) {
    // cooperative chunk load: 40 rows x 128 floats per matrix, float4 moves
    for (int idx = tid; idx < PB * (CHUNK / 4); idx += 288) {
      const int row = idx >> 5;            // idx / 32
      const int c4  = (idx & 31) * 4;
      const float4 va = *(const float4*)(gA + (size_t)row * DD + kc + c4);
      const float4 vb = *(const float4*)(gB + (size_t)row * DD + kc + c4);
      *(float4*)(&eT[row * LDSTR + c4]) = va;
      *(float4*)(&tT[row * LDSTR + c4]) = vb;
    }
    __syncthreads();

#pragma unroll 8
    for (int k0 = 0; k0 < CHUNK; k0 += 4) {
      v2f a = *(const v2f*)(aBase + k0);
      v2f b = *(const v2f*)(bBase + k0);
      // D = A(16x4 f32) * B(4x16 f32) + C(16x16 f32)
      acc = __builtin_amdgcn_wmma_f32_16x16x4_f32(
          /*neg_a=*/false, a, /*neg_b=*/false, b,
          /*c_mod=*/(short)0, acc, /*reuse_a=*/false, /*reuse_b=*/false);
    }
    __syncthreads();  // safe to overwrite LDS chunk
  }

  // Stage-1 gate + masked reduction over j. C layout: vgpr r, lanes 0-15 hold
  // row 16*tm + r (j = lane + 16*tn), lanes 16-31 hold row 16*tm + r + 8.
  const int jLocal = tn * 16 + (lane & 15);
  const bool jValid = (jLocal < PB);
#pragma unroll
  for (int r = 0; r < 8; ++r) {
    const float x = acc[r];
    float gx = 0.f, nz = 0.f;
    if (jValid) {
      nz = (x != 0.f) ? 1.f : 0.f;         // count uses PRE-gate pho
      gx = gate8(x, w1, b1, w2) * x;
    }
    // deterministic 16-lane half reduction (masks < 16 stay within the half)
#pragma unroll
    for (int m = 1; m < 16; m <<= 1) {
      gx += __shfl_xor(gx, m, 32);
      nz += __shfl_xor(nz, m, 32);
    }
    const int iLocal = tm * 16 + r + ((lane >> 4) << 3);
    if ((lane & 15) == 0 && iLocal < PB) {   // lane 0 (rows r), lane 16 (rows r+8)
      pS[iLocal][tn] = gx;
      pC[iLocal][tn] = nz;
    }
  }
  __syncthreads();

  // Stage-2: combine the 3 j-tile partials per row, second gate over P.
  if (tid < PB) {
    const float s = pS[tid][0] + pS[tid][1] + pS[tid][2];
    const float c = pC[tid][0] + pC[tid][1] + pC[tid][2];
    const float x = s / (c + 1e-6f);
    s2C[tid] = (x != 0.f) ? 1.f : 0.f;
    s2S[tid] = gate8(x, w1, b1, w2) * x;
  }
  __syncthreads();

  if (tid == 0) {
    float s = 0.f, c = 0.f;
    for (int i = 0; i < PB; ++i) { s += s2S[i]; c += s2C[i]; }
    out[blockIdx.x] = s / (c + 1e-6f);
  }
}

// ---------------------------------------------------------------------------
extern "C" void kernel_launch(void* const* d_in, const int* in_sizes, int n_in,
                              void* d_out, int out_size, void* d_ws, size_t ws_size,
                              hipStream_t stream) {
  const float* enroll = (const float*)d_in[0];   // [96,256,40]
  const float* test   = (const float*)d_in[1];   // [96,256,40]
  const float* fc1w   = (const float*)d_in[2];   // [8,1]
  const float* fc1b   = (const float*)d_in[3];   // [8]
  const float* fc2w   = (const float*)d_in[4];   // [1,8]
  float* out = (float*)d_out;                    // [96,96]

  float* Anorm = (float*)d_ws;                   // [3840][256]
  float* Bnorm = Anorm + (size_t)NROWS * DD;     // [3840][256]

  normalize_cols<<<2 * NROWS, DD, 0, stream>>>(enroll, test, Anorm, Bnorm);
  pho_fused<<<KT * KT, 288, 0, stream>>>(Anorm, Bnorm, fc1w, fc1b, fc2w, out);
}